// GAT_Encoder_89404039233750
// MI455X (gfx1250) — compile-verified
//
#include <hip/hip_runtime.h>
#include <math.h>

typedef __attribute__((ext_vector_type(2))) float v2f;
typedef __attribute__((ext_vector_type(8))) float v8f;

#define NEG_SLOPE 0.2f

__device__ __forceinline__ float lrelu(float x) { return x > 0.f ? x : NEG_SLOPE * x; }

// monotone order-preserving float <-> uint mapping (for atomicMax-based segment max)
__device__ __forceinline__ unsigned f2ord(float f) {
    unsigned u = __float_as_uint(f);
    return (u & 0x80000000u) ? ~u : (u | 0x80000000u);
}
__device__ __forceinline__ float ord2f(unsigned u) {
    return (u & 0x80000000u) ? __uint_as_float(u & 0x7FFFFFFFu) : __uint_as_float(~u);
}

__device__ __forceinline__ void atomAddF(float* p, float v) { unsafeAtomicAdd(p, v); }

// ---------------------------------------------------------------------------
// zero-fill (accumulators must be re-initialized every call)
// ---------------------------------------------------------------------------
__global__ void zero_f32(float* __restrict__ p, long n) {
    long i = (long)blockIdx.x * blockDim.x + threadIdx.x;
    if (i < n) p[i] = 0.f;
}

// ---------------------------------------------------------------------------
// Dual GEMM via V_WMMA_F32_16X16X4_F32:  O1 = A*W1, O2 = A*W2
// A: [rows,K] row-major, W: [K,Co] row-major. One wave -> one 16x16 tile of
// each output, sharing the A fragments. rows%16==0, Co%16==0, K%4==0.
// ---------------------------------------------------------------------------
__global__ void gemm_dual_wmma(const float* __restrict__ A,
                               const float* __restrict__ W1,
                               const float* __restrict__ W2,
                               float* __restrict__ O1,
                               float* __restrict__ O2,
                               int K, int Co) {
    const int rt   = blockIdx.x;       // row tile
    const int ct   = blockIdx.y;       // col tile
    const int lane = threadIdx.x;      // 0..31
    const int half = lane >> 4;        // 0: K={0,1}, 1: K={2,3} (per ISA A/B layout)
    const int lm   = lane & 15;

    v8f c1 = {}; v8f c2 = {};
    const float* Arow = A + (size_t)(rt * 16 + lm) * K + 2 * half;
    const int    col  = ct * 16 + lm;
    const float* w1p  = W1 + (size_t)(2 * half) * Co + col;
    const float* w2p  = W2 + (size_t)(2 * half) * Co + col;

    for (int k = 0; k < K; k += 4) {
        v2f a;  a.x  = Arow[k];               a.y  = Arow[k + 1];
        v2f b1; b1.x = w1p[(size_t)k * Co];   b1.y = w1p[(size_t)(k + 1) * Co];
        v2f b2; b2.x = w2p[(size_t)k * Co];   b2.y = w2p[(size_t)(k + 1) * Co];
        c1 = __builtin_amdgcn_wmma_f32_16x16x4_f32(false, a, false, b1, (short)0, c1, false, false);
        c2 = __builtin_amdgcn_wmma_f32_16x16x4_f32(false, a, false, b2, (short)0, c2, false, false);
    }

    // D layout: VGPR r -> lanes0-15: M=r, lanes16-31: M=r+8
    const size_t rb = (size_t)(rt * 16 + half * 8);
    for (int r = 0; r < 8; ++r) {
        O1[(rb + r) * Co + col] = c1[r];
        O2[(rb + r) * Co + col] = c2[r];
    }
}

// ---------------------------------------------------------------------------
// Layer 1 edge pipeline (H=2 heads x C1=64 ch = 128 floats/row)
// ---------------------------------------------------------------------------
__device__ __forceinline__ void edge_ids(const int* __restrict__ ei, int Eorig,
                                         int e, int& src, int& dst) {
    if (e < Eorig) { src = ei[e]; dst = ei[Eorig + e]; }
    else           { src = dst = e - Eorig; }   // self loops
}

__global__ void edge_logits1(const int* __restrict__ ei, int Eorig, int Etot,
                             const float* __restrict__ xl1, const float* __restrict__ xr1,
                             const float* __restrict__ att1,
                             float* __restrict__ lp1, unsigned* __restrict__ m1) {
    int wave = blockIdx.x * (blockDim.x >> 5) + (threadIdx.x >> 5);
    int lane = threadIdx.x & 31;
    if (wave >= Etot) return;
    int src, dst; edge_ids(ei, Eorig, wave, src, dst);

    float4 a = ((const float4*)(xl1 + (size_t)src * 128))[lane];
    float4 b = ((const float4*)(xr1 + (size_t)dst * 128))[lane];
    float4 t = ((const float4*)att1)[lane];   // flat [h*64+c]; lane<16 -> head0
    float s = lrelu(a.x + b.x) * t.x + lrelu(a.y + b.y) * t.y +
              lrelu(a.z + b.z) * t.z + lrelu(a.w + b.w) * t.w;
    for (int m = 1; m < 16; m <<= 1) s += __shfl_xor(s, m, 32);  // per-head halves
    if ((lane & 15) == 0) {
        int h = lane >> 4;
        lp1[(size_t)wave * 2 + h] = s;
        atomicMax(&m1[(size_t)dst * 2 + h], f2ord(s));
    }
}

__global__ void edge_softmax1(const int* __restrict__ ei, int Eorig, int Etot,
                              float* __restrict__ lp1,
                              const unsigned* __restrict__ m1, float* __restrict__ s1) {
    int i = blockIdx.x * blockDim.x + threadIdx.x;
    if (i >= 2 * Etot) return;
    int e = i >> 1, h = i & 1;
    int src, dst; edge_ids(ei, Eorig, e, src, dst); (void)src;
    float p = expf(lp1[i] - ord2f(m1[(size_t)dst * 2 + h]));
    lp1[i] = p;                                  // reuse buffer: logits -> p
    atomAddF(&s1[(size_t)dst * 2 + h], p);
}

__global__ void edge_aggregate1(const int* __restrict__ ei, int Eorig, int Etot,
                                const float* __restrict__ xl1,
                                const float* __restrict__ lp1, const float* __restrict__ s1,
                                float* __restrict__ h1) {
    int wave = blockIdx.x * (blockDim.x >> 5) + (threadIdx.x >> 5);
    int lane = threadIdx.x & 31;
    if (wave >= Etot) return;
    int src, dst; edge_ids(ei, Eorig, wave, src, dst);
    int h = lane >> 4;                           // channels lane*4.. belong to head lane/16
    float alpha = lp1[(size_t)wave * 2 + h] / s1[(size_t)dst * 2 + h];
    float4 msg = ((const float4*)(xl1 + (size_t)src * 128))[lane];
    float* o = h1 + (size_t)dst * 128 + lane * 4;
    atomAddF(o + 0, alpha * msg.x);
    atomAddF(o + 1, alpha * msg.y);
    atomAddF(o + 2, alpha * msg.z);
    atomAddF(o + 3, alpha * msg.w);
}

__global__ void elu_bias128(float* __restrict__ h1, const float* __restrict__ b1, long n) {
    long i = (long)blockIdx.x * blockDim.x + threadIdx.x;
    if (i >= n) return;
    float v = h1[i] + b1[i & 127];
    h1[i] = v > 0.f ? v : expm1f(v);
}

// ---------------------------------------------------------------------------
// Layer 2 edge pipeline (1 head x 32 ch)
// ---------------------------------------------------------------------------
__global__ void edge_logits2(const int* __restrict__ ei, int Eorig, int Etot,
                             const float* __restrict__ xl2, const float* __restrict__ xr2,
                             const float* __restrict__ att2,
                             float* __restrict__ lp2, unsigned* __restrict__ m2) {
    int wave = blockIdx.x * (blockDim.x >> 5) + (threadIdx.x >> 5);
    int lane = threadIdx.x & 31;
    if (wave >= Etot) return;
    int src, dst; edge_ids(ei, Eorig, wave, src, dst);
    float v = lrelu(xl2[(size_t)src * 32 + lane] + xr2[(size_t)dst * 32 + lane]) * att2[lane];
    for (int m = 1; m < 32; m <<= 1) v += __shfl_xor(v, m, 32);
    if (lane == 0) {
        lp2[wave] = v;
        atomicMax(&m2[dst], f2ord(v));
    }
}

__global__ void edge_softmax2(const int* __restrict__ ei, int Eorig, int Etot,
                              float* __restrict__ lp2,
                              const unsigned* __restrict__ m2, float* __restrict__ s2) {
    int e = blockIdx.x * blockDim.x + threadIdx.x;
    if (e >= Etot) return;
    int src, dst; edge_ids(ei, Eorig, e, src, dst); (void)src;
    float p = expf(lp2[e] - ord2f(m2[dst]));
    lp2[e] = p;
    atomAddF(&s2[dst], p);
}

__global__ void edge_aggregate2(const int* __restrict__ ei, int Eorig, int Etot,
                                const float* __restrict__ xl2,
                                const float* __restrict__ lp2, const float* __restrict__ s2,
                                float* __restrict__ acc2) {
    int wave = blockIdx.x * (blockDim.x >> 5) + (threadIdx.x >> 5);
    int lane = threadIdx.x & 31;
    if (wave >= Etot) return;
    int src, dst; edge_ids(ei, Eorig, wave, src, dst);
    float alpha = lp2[wave] / s2[dst];
    atomAddF(&acc2[(size_t)dst * 32 + lane], alpha * xl2[(size_t)src * 32 + lane]);
}

__global__ void final_bias32(const float* __restrict__ acc2, const float* __restrict__ b2,
                             float* __restrict__ out, long n) {
    long i = (long)blockIdx.x * blockDim.x + threadIdx.x;
    if (i >= n) return;
    out[i] = acc2[i] + b2[i & 31];
}

// ---------------------------------------------------------------------------
extern "C" void kernel_launch(void* const* d_in, const int* in_sizes, int n_in,
                              void* d_out, int out_size, void* d_ws, size_t ws_size,
                              hipStream_t stream) {
    const float* x    = (const float*)d_in[0];
    const int*   ei   = (const int*)d_in[1];
    const float* Wl1  = (const float*)d_in[2];
    const float* Wr1  = (const float*)d_in[3];
    const float* att1 = (const float*)d_in[4];
    const float* b1   = (const float*)d_in[5];
    const float* Wl2  = (const float*)d_in[6];
    const float* Wr2  = (const float*)d_in[7];
    const float* att2 = (const float*)d_in[8];
    const float* b2   = (const float*)d_in[9];
    float* out = (float*)d_out;

    const int Nn    = in_sizes[0] / 128;   // 50000
    const int Eorig = in_sizes[1] / 2;     // 800000
    const int Etot  = Eorig + Nn;          // + self loops

    // ---- workspace layout (floats) ----
    float* f = (float*)d_ws;
    const size_t N128 = (size_t)Nn * 128, N32 = (size_t)Nn * 32;
    float*    xl1  = f;                 // N*128
    float*    xr1  = xl1 + N128;        // N*128
    float*    h1   = xr1 + N128;        // N*128 accum -> (in place) elu(h+b1)
    float*    xl2  = h1 + N128;         // N*32
    float*    xr2  = xl2 + N32;         // N*32
    float*    acc2 = xr2 + N32;         // N*32 accum
    unsigned* m1   = (unsigned*)(acc2 + N32);  // 2N
    float*    s1   = (float*)(m1 + 2 * (size_t)Nn); // 2N
    unsigned* m2   = (unsigned*)(s1 + 2 * (size_t)Nn); // N
    float*    s2   = (float*)(m2 + Nn); // N
    float*    lp1  = s2 + Nn;           // 2*Etot (logits then p)
    float*    lp2  = lp1 + 2 * (size_t)Etot;   // Etot

    // ---- 0) zero accumulators: contiguous h1 .. s2 = 230*N floats ----
    {
        long nz = 230L * Nn;
        zero_f32<<<(unsigned)((nz + 255) / 256), 256, 0, stream>>>(h1, nz);
    }

    const int waveBlocks = (Etot + 7) / 8;   // 8 waves (256 thr) per block

    // ---- 1) layer-1 transforms: xl1 = x@Wl1, xr1 = x@Wr1 (WMMA f32) ----
    gemm_dual_wmma<<<dim3(Nn / 16, 128 / 16), 32, 0, stream>>>(x, Wl1, Wr1, xl1, xr1, 128, 128);

    // ---- 2) layer-1 attention ----
    edge_logits1<<<waveBlocks, 256, 0, stream>>>(ei, Eorig, Etot, xl1, xr1, att1, lp1, m1);
    edge_softmax1<<<(2 * Etot + 255) / 256, 256, 0, stream>>>(ei, Eorig, Etot, lp1, m1, s1);
    edge_aggregate1<<<waveBlocks, 256, 0, stream>>>(ei, Eorig, Etot, xl1, lp1, s1, h1);
    elu_bias128<<<(unsigned)((N128 + 255) / 256), 256, 0, stream>>>(h1, b1, (long)N128);

    // ---- 3) layer-2 transforms: xl2 = h1@Wl2, xr2 = h1@Wr2 (WMMA f32) ----
    gemm_dual_wmma<<<dim3(Nn / 16, 32 / 16), 32, 0, stream>>>(h1, Wl2, Wr2, xl2, xr2, 128, 32);

    // ---- 4) layer-2 attention ----
    edge_logits2<<<waveBlocks, 256, 0, stream>>>(ei, Eorig, Etot, xl2, xr2, att2, lp2, m2);
    edge_softmax2<<<(Etot + 255) / 256, 256, 0, stream>>>(ei, Eorig, Etot, lp2, m2, s2);
    edge_aggregate2<<<waveBlocks, 256, 0, stream>>>(ei, Eorig, Etot, xl2, lp2, s2, acc2);

    // ---- 5) bias, write output [N,32] ----
    final_bias32<<<(unsigned)((N32 + 255) / 256), 256, 0, stream>>>(acc2, b2, out, (long)N32);
}